// MixedDimTableBatchedEmbeddingBags_48567490183510
// MI455X (gfx1250) — compile-verified
//
#include <hip/hip_runtime.h>

// ---------------------------------------------------------------------------
// MixedDim table-batched embedding-bag pooling, MI455X (gfx1250).
// Memory-bound random gather (~545 MB/call -> ~23 us floor @ 23.3 TB/s).
// CDNA5 paths: GLOBAL_LOAD_ASYNC_TO_LDS_B128 (+ s_wait_asynccnt) stages bag
// metadata into LDS; weight rows gathered with NT b128 loads in SADDR form
// (SGPR base + 32-bit voffset -- all table offsets < 2^31); NT b128 stores.
// ---------------------------------------------------------------------------

typedef float v4f __attribute__((ext_vector_type(4)));

#define BATCH      8192
#define BAG_L      32
#define OUT_STRIDE 480

// Async copy of 16 bytes: global -> LDS. Tracked with ASYNCcnt.
__device__ __forceinline__ void async_copy_b128(unsigned lds_off, const void* gaddr) {
    asm volatile("global_load_async_to_lds_b128 %0, %1, off"
                 :: "v"(lds_off), "v"(gaddr)
                 : "memory");
}

__device__ __forceinline__ void wait_asynccnt0() {
    asm volatile("s_wait_asynccnt 0x0" ::: "memory");
}

// Low 32 bits of a generic pointer to __shared__ data == LDS byte offset
// (generic LDS address = {shared aperture hi, 32-bit group-segment offset}).
__device__ __forceinline__ unsigned lds_offset(const void* p) {
    return (unsigned)(unsigned long long)p;
}

template <int D>
__global__ __launch_bounds__(256)
void emb_bag_pool_kernel(const float* __restrict__ wt,   // this table's weights [250000, D]
                         const int*   __restrict__ idx,  // this table's indices [B*L]
                         const float* __restrict__ psw,  // this table's sample weights [B*L]
                         float*       __restrict__ out,  // [B, 480]
                         int col_off)                    // column offset of this table
{
    constexpr int      TPB      = D / 4;         // threads per bag (float4 each)
    constexpr int      BAGS     = 256 / TPB;     // bags per 256-thread block
    constexpr int      NSTG     = BAGS * BAG_L;  // ints (and floats) staged per block
    constexpr int      NV       = NSTG / 4;      // b128 transfers per staged array
    constexpr unsigned ROWBYTES = (unsigned)(D * sizeof(float));

    __shared__ alignas(16) int   s_idx[NSTG];
    __shared__ alignas(16) float s_psw[NSTG];

    const int       tid  = threadIdx.x;
    const long long bag0 = (long long)blockIdx.x * BAGS;

    // ---- Stage this block's bag metadata into LDS via async copies --------
    {
        const char* gi = (const char*)(idx + bag0 * BAG_L);
        const char* gp = (const char*)(psw + bag0 * BAG_L);
        const unsigned li = lds_offset(&s_idx[0]);
        const unsigned lp = lds_offset(&s_psw[0]);
        for (int v = tid; v < NV; v += 256) {
            async_copy_b128(li + v * 16, gi + v * 16);
            async_copy_b128(lp + v * 16, gp + v * 16);
        }
    }
    wait_asynccnt0();
    __syncthreads();

    // ---- Gather + weighted-sum pool ---------------------------------------
    const int bagl = tid / TPB;          // bag within block
    const int j    = tid % TPB;          // float4 column within row
    const int*   bi = &s_idx[bagl * BAG_L];
    const float* bp = &s_psw[bagl * BAG_L];

    // SGPR base (uniform) + per-thread column byte offset; per-load offset is
    // a single 32-bit mad: row * ROWBYTES + colbytes  -> SADDR-form b128 load.
    const char*    wbase    = (const char*)wt;
    const unsigned colbytes = (unsigned)j * 16u;

    v4f acc = {0.0f, 0.0f, 0.0f, 0.0f};
#pragma unroll
    for (int l = 0; l < BAG_L; ++l) {
        const unsigned row = (unsigned)bi[l];         // LDS broadcast
        const float    sw  = bp[l];                   // LDS broadcast
        const unsigned off = row * ROWBYTES + colbytes;
        const v4f* src = (const v4f*)(wbase + off);
        const v4f  val = __builtin_nontemporal_load(src);   // single-touch: NT
        acc += val * sw;
    }

    const long long b   = bag0 + bagl;
    v4f*            dst = (v4f*)(out + b * (long long)OUT_STRIDE + col_off) + j;
    __builtin_nontemporal_store(acc, dst);
}

extern "C" void kernel_launch(void* const* d_in, const int* in_sizes, int n_in,
                              void* d_out, int out_size, void* d_ws, size_t ws_size,
                              hipStream_t stream) {
    (void)in_sizes; (void)n_in; (void)out_size; (void)d_ws; (void)ws_size;

    const float* weights = (const float*)d_in[0];
    const int*   indices = (const int*)d_in[1];   // harness: integer -> const int*
    // d_in[2] = offsets: fixed-stride (L=32) encoding, not needed
    const float* psw     = (const float*)d_in[3];
    float*       out     = (float*)d_out;

    static const int dims[8] = {16, 32, 64, 128, 16, 32, 64, 128};
    const long long  rows    = 250000;

    long long w_off = 0;
    int       col   = 0;
    for (int t = 0; t < 8; ++t) {
        const int d    = dims[t];
        const int bags = 1024 / d;            // bags per block
        const int grid = BATCH / bags;        // blocks for this table
        const int*   ti = indices + (long long)t * BATCH * BAG_L;
        const float* tp = psw     + (long long)t * BATCH * BAG_L;
        const float* tw = weights + w_off;
        switch (d) {
            case 16:  emb_bag_pool_kernel<16> <<<grid, 256, 0, stream>>>(tw, ti, tp, out, col); break;
            case 32:  emb_bag_pool_kernel<32> <<<grid, 256, 0, stream>>>(tw, ti, tp, out, col); break;
            case 64:  emb_bag_pool_kernel<64> <<<grid, 256, 0, stream>>>(tw, ti, tp, out, col); break;
            case 128: emb_bag_pool_kernel<128><<<grid, 256, 0, stream>>>(tw, ti, tp, out, col); break;
        }
        w_off += rows * d;
        col   += d;
    }
}